// DetectionCriterion_63754494542413
// MI455X (gfx1250) — compile-verified
//
#include <hip/hip_runtime.h>

// Problem constants (from reference setup_inputs)
#define B_    16
#define Q_    300
#define T_    50
#define C_    80
#define H_    128
#define W_    128
#define NCLS  81                      // C_+1
#define NBT   (B_*T_)                 // 800 matched pairs
#define NBQ   (B_*Q_)                 // 4800 logit rows
#define NHM   (B_*C_*H_*W_)           // 20,971,520 heatmap elements

#define THR_HM   256
#define NBLK_HM  4096                 // 1,048,576 threads -> exactly 5 float4 each
#define WAVES_CE 8
#define NBLK_CE  (NBQ / WAVES_CE)     // 600

typedef float v2f __attribute__((ext_vector_type(2)));
typedef float v8f __attribute__((ext_vector_type(8)));

// ---- workspace layout (float offsets) ----
#define OFF_HM    0                         // [NBLK_HM] heatmap base partials
#define OFF_CEN   (OFF_HM + NBLK_HM)        // [NBLK_CE] ce numerator partials
#define OFF_CED   (OFF_CEN + NBLK_CE)       // [NBLK_CE] ce denominator partials
#define OFF_SC    (OFF_CED + NBLK_CE)       // [8] scalars: 0 bbox,1 giou,2 corr,3 boxL1,4 numpos
#define OFF_BOXN  (OFF_SC + 8)              // [NBT*4] boxes_n (unpermuted, cxcywh normalized)
#define OFF_MTGT  (OFF_BOXN + NBT*4)        // [NBT*4] matched target boxes (cxcywh)
#define OFF_LF    (OFF_MTGT + NBT*4)        // [NBT] labels (int)
#define OFF_GX    (OFF_LF + NBT)            // [NBT] grid x (int)
#define OFF_GY    (OFF_GX + NBT)            // [NBT] grid y (int)
#define OFF_SQ    (OFF_GY + NBT)            // [NBT] src query index (int)

__device__ __forceinline__ float wave_sum32(float v) {
  #pragma unroll
  for (int o = 16; o > 0; o >>= 1) v += __shfl_xor(v, o, 32);
  return v;
}
__device__ __forceinline__ float wave_max32(float v) {
  #pragma unroll
  for (int o = 16; o > 0; o >>= 1) v = fmaxf(v, __shfl_xor(v, o, 32));
  return v;
}

// focal term for target==0: 0.75 * sigmoid(x)^2 * softplus(x)
__device__ __forceinline__ float focal_neg(float x) {
  float p  = 1.0f / (1.0f + __builtin_expf(-x));
  float sp = fmaxf(x, 0.0f) + log1pf(__builtin_expf(-fabsf(x)));
  return 0.75f * p * p * sp;
}

// ---------------- prep: per (b,t) matched data ----------------
__global__ void prep_kernel(const float* __restrict__ tgt_boxes,
                            const int*   __restrict__ tgt_labels,
                            const float* __restrict__ tgt_sizes,
                            const int*   __restrict__ src_idx,
                            const int*   __restrict__ tgt_idx,
                            float* ws) {
  int i = blockIdx.x * blockDim.x + threadIdx.x;
  if (i >= NBT) return;
  int b = i / T_, t = i - b * T_;
  float him = tgt_sizes[b * 2 + 0], wim = tgt_sizes[b * 2 + 1];
  // boxes_n[i]: normalize then xyxy->cxcywh (order commutes componentwise)
  const float* bx = tgt_boxes + (b * T_ + t) * 4;
  float cx = (bx[0] + bx[2]) * 0.5f / wim;
  float cy = (bx[1] + bx[3]) * 0.5f / him;
  float w  = (bx[2] - bx[0]) / wim;
  float h  = (bx[3] - bx[1]) / him;
  float* bn = ws + OFF_BOXN + i * 4;
  bn[0] = cx; bn[1] = cy; bn[2] = w; bn[3] = h;

  int* lf = (int*)(ws + OFF_LF);
  int* gx = (int*)(ws + OFF_GX);
  int* gy = (int*)(ws + OFF_GY);
  int* sq = (int*)(ws + OFF_SQ);
  lf[i] = tgt_labels[b * T_ + t];
  int gxi = (int)(cx * (float)W_); gxi = min(max(gxi, 0), W_ - 1);
  int gyi = (int)(cy * (float)H_); gyi = min(max(gyi, 0), H_ - 1);
  gx[i] = gxi; gy[i] = gyi;
  sq[i] = src_idx[b * T_ + t];

  // matched_tgt[i] = boxes_n[b, tgt_idx[b,t]]
  int ti = tgt_idx[b * T_ + t];
  const float* b2 = tgt_boxes + (b * T_ + ti) * 4;
  float* mt = ws + OFF_MTGT + i * 4;
  mt[0] = (b2[0] + b2[2]) * 0.5f / wim;
  mt[1] = (b2[1] + b2[3]) * 0.5f / him;
  mt[2] = (b2[2] - b2[0]) / wim;
  mt[3] = (b2[3] - b2[1]) / him;
}

// ---------------- CE: one wave per (b,q) row of 81 logits ----------------
__global__ void __launch_bounds__(WAVES_CE * 32)
ce_kernel(const float* __restrict__ logits,
          const int*   __restrict__ tgt_labels,
          const int*   __restrict__ src_idx,
          const int*   __restrict__ tgt_idx,
          const float* __restrict__ empty_weight,
          float* ws) {
  __shared__ float snum[WAVES_CE], sden[WAVES_CE];
  int wave = threadIdx.x >> 5, lane = threadIdx.x & 31;
  int row  = blockIdx.x * WAVES_CE + wave;      // always < NBQ (4800 = 600*8)
  int b = row / Q_, q = row - b * Q_;

  // target class: scan this batch's 50 matches (scalar-friendly, uniform)
  int tc = C_;
  for (int t = 0; t < T_; ++t)
    if (src_idx[b * T_ + t] == q) tc = tgt_labels[b * T_ + tgt_idx[b * T_ + t]];

  const float* x = logits + (long)row * NCLS;
  float x0 = x[lane];
  float x1 = x[lane + 32];
  bool  h2 = lane < (NCLS - 64);                 // lanes 0..16 own a 3rd class
  float x2 = h2 ? x[lane + 64] : -3.0e38f;

  float m = wave_max32(fmaxf(fmaxf(x0, x1), x2));
  float se = __builtin_expf(x0 - m) + __builtin_expf(x1 - m) +
             (h2 ? __builtin_expf(x2 - m) : 0.0f);
  se = wave_sum32(se);
  float sel = (lane == tc ? x0 : 0.0f) + (lane + 32 == tc ? x1 : 0.0f) +
              ((h2 && lane + 64 == tc) ? x2 : 0.0f);
  sel = wave_sum32(sel);

  if (lane == 0) {
    float nll = m + __builtin_logf(se) - sel;    // -log_softmax at tc
    float cw  = empty_weight[tc];
    snum[wave] = cw * nll;
    sden[wave] = cw;
  }
  __syncthreads();
  if (threadIdx.x == 0) {
    float n = 0.f, d = 0.f;
    #pragma unroll
    for (int w2 = 0; w2 < WAVES_CE; ++w2) { n += snum[w2]; d += sden[w2]; }
    ws[OFF_CEN + blockIdx.x] = n;
    ws[OFF_CED + blockIdx.x] = d;
  }
}

// ---------------- bbox L1 + GIoU over 800 matched pairs ----------------
__global__ void box_kernel(const float* __restrict__ pred_boxes, float* ws) {
  __shared__ float s1[1024], s2[1024];
  int i = threadIdx.x;
  float l1 = 0.f, lg = 0.f;
  if (i < NBT) {
    int b = i / T_;
    const int* sq = (const int*)(ws + OFF_SQ);
    const float* s = pred_boxes + (b * Q_ + sq[i]) * 4;
    const float* t = ws + OFF_MTGT + i * 4;
    float scx = s[0], scy = s[1], sw = s[2], sh = s[3];
    float tcx = t[0], tcy = t[1], tw = t[2], th = t[3];
    l1 = fabsf(scx - tcx) + fabsf(scy - tcy) + fabsf(sw - tw) + fabsf(sh - th);
    float ax1 = scx - 0.5f * sw, ay1 = scy - 0.5f * sh, ax2 = scx + 0.5f * sw, ay2 = scy + 0.5f * sh;
    float bx1 = tcx - 0.5f * tw, by1 = tcy - 0.5f * th, bx2 = tcx + 0.5f * tw, by2 = tcy + 0.5f * th;
    float aA = (ax2 - ax1) * (ay2 - ay1), aB = (bx2 - bx1) * (by2 - by1);
    float iw = fmaxf(fminf(ax2, bx2) - fmaxf(ax1, bx1), 0.f);
    float ih = fmaxf(fminf(ay2, by2) - fmaxf(ay1, by1), 0.f);
    float inter = iw * ih, uni = aA + aB - inter;
    float iou = inter / uni;
    float cw = fmaxf(fmaxf(ax2, bx2) - fminf(ax1, bx1), 0.f);
    float ch = fmaxf(fmaxf(ay2, by2) - fminf(ay1, by1), 0.f);
    float aC = cw * ch;
    lg = 1.0f - (iou - (aC - uni) / aC);
  }
  s1[i] = l1; s2[i] = lg;
  __syncthreads();
  for (int o = 512; o > 0; o >>= 1) {
    if (i < o) { s1[i] += s1[i + o]; s2[i] += s2[i + o]; }
    __syncthreads();
  }
  if (i == 0) { ws[OFF_SC + 0] = s1[0]; ws[OFF_SC + 1] = s2[0]; }
}

// ---------------- sparse correction: focal@positives, num_pos, box L1 ----------------
__global__ void corr_kernel(const float* __restrict__ hm,
                            const float* __restrict__ box_map, float* ws) {
  __shared__ float sc[1024], sb[1024];
  __shared__ int   sp[1024];
  int i = threadIdx.x;
  float corr = 0.f, bsum = 0.f; int npos = 0;
  const int* lf = (const int*)(ws + OFF_LF);
  const int* gx = (const int*)(ws + OFF_GX);
  const int* gy = (const int*)(ws + OFF_GY);
  if (i < NBT) {
    int b = i / T_;
    int li = lf[i], xi = gx[i], yi = gy[i];
    // heatmap: apply correction once per distinct (b, label, gy, gx)
    bool firstHM = true;
    for (int j = 0; j < i; ++j)
      if (j / T_ == b && lf[j] == li && gx[j] == xi && gy[j] == yi) { firstHM = false; break; }
    if (firstHM) {
      float x = hm[((b * C_ + li) * H_ + yi) * W_ + xi];
      float p   = 1.0f / (1.0f + __builtin_expf(-x));
      float l1p = log1pf(__builtin_expf(-fabsf(x)));
      float sp_pos = fmaxf(x, 0.f) + l1p;   // softplus(x)  = ce(target 0)
      float sp_neg = fmaxf(-x, 0.f) + l1p;  // softplus(-x) = ce(target 1)
      corr = 0.25f * (1.f - p) * (1.f - p) * sp_neg - 0.75f * p * p * sp_pos;
    }
    // position mask: distinct (b, gy, gx); box scatter last-write-wins
    bool firstPos = true, lastPos = true;
    for (int j = 0; j < NBT; ++j) {
      if (j == i) continue;
      if (j / T_ == b && gx[j] == xi && gy[j] == yi) {
        if (j < i) firstPos = false; else lastPos = false;
      }
    }
    if (firstPos) npos = 1;
    if (lastPos) {
      const float* bn = ws + OFF_BOXN + i * 4;
      #pragma unroll
      for (int c = 0; c < 4; ++c)
        bsum += fabsf(box_map[((b * 4 + c) * H_ + yi) * W_ + xi] - bn[c]);
    }
  }
  sc[i] = corr; sb[i] = bsum; sp[i] = npos;
  __syncthreads();
  for (int o = 512; o > 0; o >>= 1) {
    if (i < o) { sc[i] += sc[i + o]; sb[i] += sb[i + o]; sp[i] += sp[i + o]; }
    __syncthreads();
  }
  if (i == 0) {
    ws[OFF_SC + 2] = sc[0];
    ws[OFF_SC + 3] = sb[0];
    ws[OFF_SC + 4] = (float)sp[0];
  }
}

// ---------------- main 84MB stream: sum focal(x, target=0); WMMA wave reduce ----------------
__global__ void __launch_bounds__(THR_HM)
hm_base_kernel(const float4* __restrict__ hm4, float* ws) {
  float a0 = 0.f, a1 = 0.f;
  int tid = blockIdx.x * THR_HM + threadIdx.x;
  const int n4 = NHM / 4;
  for (int idx = tid; idx < n4; idx += NBLK_HM * THR_HM) {
    float4 v = hm4[idx];
    a0 += focal_neg(v.x) + focal_neg(v.y);
    a1 += focal_neg(v.z) + focal_neg(v.w);
  }
  // Wave reduction through the matrix unit: A = 16x4 f32 (2 regs/lane = our
  // partials), B = ones(4x16)  =>  D[m][n] = rowsum(A[m]).  Wave total =
  // sum_m rowsum(m) = (lane0 half) + (lane16 half) of the per-lane D sums.
  v2f A;  A[0] = a0;  A[1] = a1;
  v2f Bv; Bv[0] = 1.0f; Bv[1] = 1.0f;
  v8f Cc = {};
  v8f D = __builtin_amdgcn_wmma_f32_16x16x4_f32(false, A, false, Bv,
                                                (short)0, Cc, false, false);
  float s = D[0] + D[1] + D[2] + D[3] + D[4] + D[5] + D[6] + D[7];
  float tot = __shfl(s, 0, 32) + __shfl(s, 16, 32);

  __shared__ float sw[THR_HM / 32];
  int wave = threadIdx.x >> 5, lane = threadIdx.x & 31;
  if (lane == 0) sw[wave] = tot;
  __syncthreads();
  if (threadIdx.x == 0) {
    float t = 0.f;
    #pragma unroll
    for (int w2 = 0; w2 < THR_HM / 32; ++w2) t += sw[w2];
    ws[OFF_HM + blockIdx.x] = t;
  }
}

// ---------------- deterministic finalize ----------------
__global__ void finalize_kernel(const float* __restrict__ ws, float* __restrict__ out) {
  __shared__ float s[256];
  int i = threadIdx.x;
  float v = 0.f;
  for (int k = i; k < NBLK_HM; k += 256) v += ws[OFF_HM + k];
  s[i] = v; __syncthreads();
  for (int o = 128; o > 0; o >>= 1) { if (i < o) s[i] += s[i + o]; __syncthreads(); }
  float hm_base = s[0]; __syncthreads();

  v = 0.f;
  for (int k = i; k < NBLK_CE; k += 256) v += ws[OFF_CEN + k];
  s[i] = v; __syncthreads();
  for (int o = 128; o > 0; o >>= 1) { if (i < o) s[i] += s[i + o]; __syncthreads(); }
  float cen = s[0]; __syncthreads();

  v = 0.f;
  for (int k = i; k < NBLK_CE; k += 256) v += ws[OFF_CED + k];
  s[i] = v; __syncthreads();
  for (int o = 128; o > 0; o >>= 1) { if (i < o) s[i] += s[i + o]; __syncthreads(); }
  float ced = s[0];

  if (i == 0) {
    float loss_ce   = cen / ced;
    float loss_bbox = ws[OFF_SC + 0] / (float)NBT;
    float loss_giou = ws[OFF_SC + 1] / (float)NBT;
    float npos      = fmaxf(ws[OFF_SC + 4], 1.0f);
    float hm_loss   = (hm_base + ws[OFF_SC + 2]) / npos;
    float box_loss  = ws[OFF_SC + 3] / npos;
    float loss_aux  = 1.0f * hm_loss + 5.0f * box_loss;
    float total = 1.0f * loss_ce + 5.0f * loss_bbox + 2.0f * loss_giou + 1.0f * loss_aux;
    out[0] = loss_ce; out[1] = loss_bbox; out[2] = loss_giou;
    out[3] = loss_aux; out[4] = total;
  }
}

extern "C" void kernel_launch(void* const* d_in, const int* in_sizes, int n_in,
                              void* d_out, int out_size, void* d_ws, size_t ws_size,
                              hipStream_t stream) {
  (void)in_sizes; (void)n_in; (void)out_size; (void)ws_size;
  const float* pred_logits = (const float*)d_in[0];
  const float* pred_boxes  = (const float*)d_in[1];
  const float* heatmap     = (const float*)d_in[2];
  const float* box_map     = (const float*)d_in[3];
  const float* tgt_boxes   = (const float*)d_in[4];
  const int*   tgt_labels  = (const int*)d_in[5];
  const float* tgt_sizes   = (const float*)d_in[6];
  const int*   src_idx     = (const int*)d_in[7];
  const int*   tgt_idx     = (const int*)d_in[8];
  const float* empty_w     = (const float*)d_in[9];
  float* ws  = (float*)d_ws;
  float* out = (float*)d_out;

  prep_kernel<<<(NBT + 255) / 256, 256, 0, stream>>>(tgt_boxes, tgt_labels, tgt_sizes,
                                                     src_idx, tgt_idx, ws);
  ce_kernel<<<NBLK_CE, WAVES_CE * 32, 0, stream>>>(pred_logits, tgt_labels, src_idx,
                                                   tgt_idx, empty_w, ws);
  box_kernel<<<1, 1024, 0, stream>>>(pred_boxes, ws);
  corr_kernel<<<1, 1024, 0, stream>>>(heatmap, box_map, ws);
  hm_base_kernel<<<NBLK_HM, THR_HM, 0, stream>>>((const float4*)heatmap, ws);
  finalize_kernel<<<1, 256, 0, stream>>>(ws, out);
}